// Yolo3_72791105733182
// MI455X (gfx1250) — compile-verified
//
#include <hip/hip_runtime.h>
#include <cstdint>

#define NCLS 80
#define PTOT 3549   // 169 + 676 + 2704
#define BATCH 64

// rcp + one Newton-Raphson step -> ~full f32 precision, far cheaper than
// clang's precise-division expansion.
__device__ __forceinline__ float fast_recip(float d) {
    float r = __builtin_amdgcn_rcpf(d);
    return r * __builtin_fmaf(-d, r, 2.0f);
}

__device__ __forceinline__ float fast_sigmoid(float x) {
    return fast_recip(1.0f + __expf(-x));
}

template <int GRID>
__global__ __launch_bounds__(128) void yolo_decode(
    const float* __restrict__ feat,
    float* __restrict__ out_pred,
    float* __restrict__ out_conf,
    float* __restrict__ out_prob,
    int posOffset,
    float aw0, float ah0, float aw1, float ah1, float aw2, float ah2)
{
    constexpr int GG    = GRID * GRID;
    constexpr int TILES = (GG + 31) / 32;

    // [wave][channel k][lane] staging: 85 rows x 32 lanes per wave
    __shared__ float tile[4][85 * 32];

    const int wid  = threadIdx.x >> 5;
    const int lane = threadIdx.x & 31;
    const int wglb = blockIdx.x * 4 + wid;

    const int tileIdx = wglb % TILES;
    int rest = wglb / TILES;
    const int a = rest % 3;
    const int b = rest / 3;
    if (b >= BATCH) return;          // uniform across the wave

    const int  p     = tileIdx * 32 + lane;
    const bool valid = (p < GG);
    const int  pc    = valid ? p : (GG - 1);   // clamp so all lanes stay in-bounds

    // Global source: feat[b, a*85 + k, y, x]; lanes span consecutive x -> coalesced
    const float* ga = feat + ((size_t)(b * 255 + a * 85) * GG + pc);

    // LDS byte address of this lane's slot in row k=0.
    // Low 32 bits of a flat shared-memory pointer == LDS byte offset (aperture rule).
    unsigned ldsa = (unsigned)(uintptr_t)(&tile[wid][lane]);

    // Async global -> LDS staging of the 85x32 tile (ASYNCcnt path, CDNA5)
    #pragma unroll
    for (int k = 0; k < 85; ++k) {
        asm volatile("global_load_async_to_lds_b32 %0, %1, off"
                     :: "v"(ldsa), "v"(ga)
                     : "memory");
        ldsa += 32 * 4;   // next channel row in LDS
        ga   += GG;       // next channel plane in global
    }

    // Async loads retire in order: <=63 outstanding guarantees the first 22
    // rows (box channels 0..4 included) are in LDS. Decode box/conf while the
    // class rows keep streaming.
    asm volatile("s_wait_asynccnt 0x3f" ::: "memory");

    const float* my = &tile[wid][lane];   // stride 32 floats between channels

    const float inv_grid = 1.0f / (float)GRID;
    const float xs = (float)(pc % GRID);
    const float ys = (float)(pc / GRID);
    const float aw = (a == 0) ? aw0 : ((a == 1) ? aw1 : aw2);
    const float ah = (a == 0) ? ah0 : ((a == 1) ? ah1 : ah2);

    // --- box decode (channels 0..4) ---
    const float s0 = fast_sigmoid(my[0 * 32]);
    const float s1 = fast_sigmoid(my[1 * 32]);
    const float e2 = __expf(my[2 * 32]);
    const float e3 = __expf(my[3 * 32]);
    const float cf = fast_sigmoid(my[4 * 32]);

    float4 pred;
    pred.x = (s0 + xs) * inv_grid;
    pred.y = (s1 + ys) * inv_grid;
    pred.z = e2 * aw * inv_grid;
    pred.w = e3 * ah * inv_grid;

    // Now require the remaining class rows.
    asm volatile("s_wait_asynccnt 0x0" ::: "memory");

    // --- softmax over 80 classes (channels 5..84) ---
    float v[NCLS];
    #pragma unroll
    for (int c = 0; c < NCLS; ++c) v[c] = my[(5 + c) * 32];

    float m = v[0];
    #pragma unroll
    for (int c = 1; c < NCLS; ++c) m = fmaxf(m, v[c]);

    float s = 0.0f;
    #pragma unroll
    for (int c = 0; c < NCLS; ++c) { v[c] = __expf(v[c] - m); s += v[c]; }
    const float r = fast_recip(s);

    if (valid) {
        const int    P    = posOffset + p;
        const size_t item = ((size_t)b * PTOT + P) * 3 + a;

        *(float4*)(out_pred + item * 4) = pred;       // 16B-aligned
        out_conf[item] = cf;

        float* op = out_prob + item * (size_t)NCLS;   // 320B-aligned
        #pragma unroll
        for (int g = 0; g < NCLS / 4; ++g) {
            float4 q;
            q.x = v[4 * g + 0] * r;
            q.y = v[4 * g + 1] * r;
            q.z = v[4 * g + 2] * r;
            q.w = v[4 * g + 3] * r;
            *(float4*)(op + 4 * g) = q;               // b128 stores
        }
    }
}

extern "C" void kernel_launch(void* const* d_in, const int* in_sizes, int n_in,
                              void* d_out, int out_size, void* d_ws, size_t ws_size,
                              hipStream_t stream) {
    (void)in_sizes; (void)n_in; (void)out_size; (void)d_ws; (void)ws_size;

    const float* feat13 = (const float*)d_in[0];
    const float* feat26 = (const float*)d_in[1];
    const float* feat52 = (const float*)d_in[2];

    float* out_pred = (float*)d_out;
    float* out_conf = out_pred + (size_t)BATCH * PTOT * 3 * 4;  // 2,725,632
    float* out_prob = out_conf + (size_t)BATCH * PTOT * 3;      // +681,408

    // waves = BATCH * 3 * ceil(GG/32); 4 waves per 128-thread block
    // GRID=13: tiles=6  -> 1152 waves -> 288 blocks
    // GRID=26: tiles=22 -> 4224 waves -> 1056 blocks
    // GRID=52: tiles=85 -> 16320 waves -> 4080 blocks
    yolo_decode<13><<<288, 128, 0, stream>>>(
        feat13, out_pred, out_conf, out_prob, /*posOffset=*/0,
        116.0f / 32.0f,  90.0f / 32.0f,
        156.0f / 32.0f, 198.0f / 32.0f,
        373.0f / 32.0f, 326.0f / 32.0f);

    yolo_decode<26><<<1056, 128, 0, stream>>>(
        feat26, out_pred, out_conf, out_prob, /*posOffset=*/169,
         30.0f / 16.0f,  61.0f / 16.0f,
         62.0f / 16.0f,  45.0f / 16.0f,
         59.0f / 16.0f, 119.0f / 16.0f);

    yolo_decode<52><<<4080, 128, 0, stream>>>(
        feat52, out_pred, out_conf, out_prob, /*posOffset=*/845,
         10.0f / 8.0f, 13.0f / 8.0f,
         16.0f / 8.0f, 30.0f / 8.0f,
         33.0f / 8.0f, 23.0f / 8.0f);
}